// GraphSAGERR_44461501448671
// MI455X (gfx1250) — compile-verified
//
#include <hip/hip_runtime.h>

// GraphSAGE 2-layer (mean aggr) for MI455X / gfx1250.
// scatter-add (f32 atomics, L2-resident) -> fused mean@Wl + b + x@Wr GEMM
// using V_WMMA_F32_16X16X4_F32 with LDS-staged A tiles (bank-conflict-free,
// 132-float pitch) and pre-packed B operands (fully coalesced b64 loads).

#define N_NODES 100000
#define N_EDGES 1600000
#define DIM     128
#define ORIG    90000
#define APITCH  132   // LDS row pitch: 132 mod 64 banks = 4 -> conflict-free

typedef __attribute__((ext_vector_type(2))) float v2f;
typedef __attribute__((ext_vector_type(8))) float v8f;

__global__ void zero_kernel(float* __restrict__ p, long n) {
    long i = (long)blockIdx.x * blockDim.x + threadIdx.x;
    long stride = (long)gridDim.x * blockDim.x;
    for (; i < n; i += stride) p[i] = 0.0f;
}

__global__ void count_kernel(const int* __restrict__ dst, float* __restrict__ cnt) {
    int e = blockIdx.x * blockDim.x + threadIdx.x;
    if (e < N_EDGES) atomicAdd(&cnt[dst[e]], 1.0f);
}

__global__ void inv_kernel(float* __restrict__ cnt) {
    int i = blockIdx.x * blockDim.x + threadIdx.x;
    if (i < N_NODES) cnt[i] = 1.0f / fmaxf(cnt[i], 1.0f);
}

// One wave per edge: float4 gather of source row (coalesced), 4 f32 atomics
// per lane into destination row. Whole agg buffer lives in the 192MB L2.
__global__ __launch_bounds__(256)
void scatter_kernel(const float* __restrict__ feat,
                    const int* __restrict__ src,
                    const int* __restrict__ dst,
                    float* __restrict__ agg) {
    int gid  = blockIdx.x * blockDim.x + threadIdx.x;
    int e    = gid >> 5;
    int lane = gid & 31;
    if (e >= N_EDGES) return;
    int s = src[e];
    int d = dst[e];
    const float4 v = *reinterpret_cast<const float4*>(feat + (size_t)s * DIM + lane * 4);
    float* b = agg + (size_t)d * DIM + lane * 4;
    atomicAdd(b + 0, v.x);
    atomicAdd(b + 1, v.y);
    atomicAdd(b + 2, v.z);
    atomicAdd(b + 3, v.w);
}

// Pack a 128x128 row-major weight matrix into the per-lane WMMA B layout:
// pair (W[4kk+2h][c], W[4kk+2h+1][c]) stored at Wp[((kk*2+h)*128 + c)*2].
// Lane l of wave w then reads an 8B v2f at ((kk*2 + (l>>4))*128 + w*16 +
// (l&15))*2 -> lanes 0-15 / 16-31 each hit one contiguous 128B segment.
__global__ __launch_bounds__(256)
void pack_weights_kernel(const float* __restrict__ W, float* __restrict__ Wp) {
    int tid = blockIdx.x * blockDim.x + threadIdx.x;   // 0 .. 8191 (pairs)
    if (tid >= (DIM * DIM) / 2) return;
    int kk  = tid >> 8;            // /256: k-step (0..31)
    int rem = tid & 255;
    int h   = rem >> 7;            // 0/1: lane-half
    int c   = rem & 127;           // column
    int krow = kk * 4 + h * 2;
    Wp[tid * 2 + 0] = W[(krow + 0) * DIM + c];
    Wp[tid * 2 + 1] = W[(krow + 1) * DIM + c];
}

// Fused SAGE layer: out[16 rows x 128 cols per block] =
//   (agg*inv) @ Wl + bl + xin @ Wr, via V_WMMA_F32_16X16X4_F32.
// A tiles (mean, x) staged in LDS with coalesced float4 loads; inv applied
// during staging. 8 waves per block, one 16x16 tile per wave.
__global__ __launch_bounds__(256)
void sage_gemm_kernel(const float* __restrict__ agg,
                      const float* __restrict__ inv,
                      const float* __restrict__ xin,
                      const float* __restrict__ Wlp,   // packed Wl
                      const float* __restrict__ bl,
                      const float* __restrict__ Wrp,   // packed Wr
                      float* __restrict__ out) {
    __shared__ float sMean[16 * APITCH];
    __shared__ float sX[16 * APITCH];

    const int tid     = threadIdx.x;
    const int rowBase = blockIdx.x * 16;

    // Stage 16x128 tiles: 512 float4 per tile, 2 per thread, coalesced.
#pragma unroll
    for (int i = 0; i < 2; i++) {
        int idx = tid + i * 256;      // 0..511
        int rr  = idx >> 5;           // row 0..15
        int c4  = idx & 31;           // float4 column 0..31
        const size_t goff = (size_t)(rowBase + rr) * DIM + c4 * 4;
        const float  iv   = inv[rowBase + rr];
        float4 va = *reinterpret_cast<const float4*>(agg + goff);
        va.x *= iv; va.y *= iv; va.z *= iv; va.w *= iv;
        *reinterpret_cast<float4*>(&sMean[rr * APITCH + c4 * 4]) = va;
        *reinterpret_cast<float4*>(&sX[rr * APITCH + c4 * 4]) =
            *reinterpret_cast<const float4*>(xin + goff);
    }
    __syncthreads();

    const int lane    = tid & 31;
    const int wave    = tid >> 5;
    const int n16     = lane & 15;       // M for A-operand, N for B/C/D
    const int half    = lane >> 4;       // K-pair select
    const int khalf   = half << 1;
    const int colBase = wave * 16;

    const float* aLds = &sMean[n16 * APITCH + khalf];
    const float* xLds = &sX[n16 * APITCH + khalf];
    const float* blp  = Wlp + ((size_t)half * DIM + colBase + n16) * 2;
    const float* brp  = Wrp + ((size_t)half * DIM + colBase + n16) * 2;

    v8f acc = {0.f, 0.f, 0.f, 0.f, 0.f, 0.f, 0.f, 0.f};

#pragma unroll 4
    for (int kk = 0; kk < DIM / 4; kk++) {
        // ---- mean @ Wl ----
        v2f a = *reinterpret_cast<const v2f*>(aLds + kk * 4);        // ds_load_b64
        v2f b = *reinterpret_cast<const v2f*>(blp + kk * 512);       // coalesced b64
        acc = __builtin_amdgcn_wmma_f32_16x16x4_f32(false, a, false, b,
                                                    (short)0, acc, false, false);
        // ---- x @ Wr ----
        a = *reinterpret_cast<const v2f*>(xLds + kk * 4);
        b = *reinterpret_cast<const v2f*>(brp + kk * 512);
        acc = __builtin_amdgcn_wmma_f32_16x16x4_f32(false, a, false, b,
                                                    (short)0, acc, false, false);
    }

    const float bias  = bl[colBase + n16];
    const int   mBase = rowBase + (half << 3);   // lanes 16-31 hold M=8..15
#pragma unroll
    for (int v = 0; v < 8; v++) {
        out[(size_t)(mBase + v) * DIM + colBase + n16] = acc[v] + bias;
    }
}

extern "C" void kernel_launch(void* const* d_in, const int* in_sizes, int n_in,
                              void* d_out, int out_size, void* d_ws, size_t ws_size,
                              hipStream_t stream) {
    const float* x   = (const float*)d_in[0];
    const int*   ei  = (const int*)d_in[1];
    const int*   src = ei;
    const int*   dst = ei + N_EDGES;
    // d_in[2] = original_size scalar (compile-time ORIG used for grid sizing)
    const float* Wl1 = (const float*)d_in[3];
    const float* bl1 = (const float*)d_in[4];
    const float* Wr1 = (const float*)d_in[5];
    const float* Wl2 = (const float*)d_in[6];
    const float* bl2 = (const float*)d_in[7];
    const float* Wr2 = (const float*)d_in[8];
    float*       out = (float*)d_out;

    // Workspace: agg [N*128] | h [N*128] | cnt/inv [N] | 4 packed weights
    float* agg  = (float*)d_ws;
    float* h    = agg + (size_t)N_NODES * DIM;
    float* cnt  = h + (size_t)N_NODES * DIM;
    float* Wp1l = cnt + N_NODES;
    float* Wp1r = Wp1l + DIM * DIM;
    float* Wp2l = Wp1r + DIM * DIM;
    float* Wp2r = Wp2l + DIM * DIM;

    const long aggN = (long)N_NODES * DIM;
    const int  packBlocks = (DIM * DIM / 2 + 255) / 256;   // 32

    zero_kernel<<<2048, 256, 0, stream>>>(agg, aggN);
    zero_kernel<<<128, 256, 0, stream>>>(cnt, (long)N_NODES);

    pack_weights_kernel<<<packBlocks, 256, 0, stream>>>(Wl1, Wp1l);
    pack_weights_kernel<<<packBlocks, 256, 0, stream>>>(Wr1, Wp1r);
    pack_weights_kernel<<<packBlocks, 256, 0, stream>>>(Wl2, Wp2l);
    pack_weights_kernel<<<packBlocks, 256, 0, stream>>>(Wr2, Wp2r);

    count_kernel<<<(N_EDGES + 255) / 256, 256, 0, stream>>>(dst, cnt);
    scatter_kernel<<<(N_EDGES * 32) / 256, 256, 0, stream>>>(x, src, dst, agg);
    inv_kernel<<<(N_NODES + 255) / 256, 256, 0, stream>>>(cnt);

    // Layer 1: h = mean@Wl1 + bl1 + x@Wr1 (all N rows)
    sage_gemm_kernel<<<N_NODES / 16, 256, 0, stream>>>(agg, cnt, x, Wp1l, bl1, Wp1r, h);

    // Layer 2 aggregation over h
    zero_kernel<<<2048, 256, 0, stream>>>(agg, aggN);
    scatter_kernel<<<(N_EDGES * 32) / 256, 256, 0, stream>>>(h, src, dst, agg);

    // Layer 2 GEMM: only first ORIG rows written to d_out (ORIG % 16 == 0)
    sage_gemm_kernel<<<ORIG / 16, 256, 0, stream>>>(agg, cnt, h, Wp2l, bl2, Wp2r, out);
}